// Mix_ScaledDotProductAttention_32555852104057
// MI455X (gfx1250) — compile-verified
//
#include <hip/hip_runtime.h>
#include <stdint.h>

// MI455X / gfx1250 dual-masked attention (causal heads 0-7, anti-causal 8-15)
// materializing full softmax weights (537 MB -> HBM-write bound, ~26us at
// 23.3 TB/s). Matmuls via V_WMMA_F32_16X16X4_F32 (f32-faithful CDNA5 matrix
// path). Two-pass flash-style softmax: pass1 online per-lane max/sum, pass2
// recompute + write weights (b128 coalesced via LDS) + accumulate O = W @ V
// with the W tile relaid C-layout -> A-layout through per-wave LDS
// (same-wave DS ops are in-order, so no barrier needed).

typedef __attribute__((ext_vector_type(2))) float v2f;
typedef __attribute__((ext_vector_type(4))) float v4f;
typedef __attribute__((ext_vector_type(8))) float v8f;

#define B_   2
#define H_   16
#define L_   2048
#define E_   64
#define NTJ  (L_ / 16)   // 128 key tiles
#define WAVES 4          // waves per block (wave32)
#define PITCH 20         // LDS row pitch (floats): %4==0 -> b64/b128 aligned

__global__ __launch_bounds__(32 * WAVES)
void attn_dualmask_wmma(const float* __restrict__ q,
                        const float* __restrict__ k,      // (B,H,E,L)
                        const float* __restrict__ v,
                        const unsigned char* __restrict__ amask, // (B,L,L)
                        const float* __restrict__ scale_p,
                        float* __restrict__ out,          // (B,H,L,E)
                        float* __restrict__ wts)          // (B,H,L,L)
{
    __shared__ float wlds[WAVES][16 * PITCH];

    const int tid  = threadIdx.x;
    const int wave = tid >> 5;
    const int lane = tid & 31;
    const int half = lane >> 4;   // which 16-lane half of the wave
    const int n    = lane & 15;   // column within a 16-wide fragment

    int task = blockIdx.x * WAVES + wave;        // 0 .. 4095
    const int qb = task % NTJ;  task /= NTJ;     // query tile (16 rows)
    const int h  = task % H_;   task /= H_;
    const int b  = task;
    const bool causal = (h < (H_ / 2));
    const int  csign  = causal ? 1 : -1;         // wave-uniform, branchless tri
    const int qrow0 = qb * 16;
    const float scale = scale_p[0];

    const float* qptr = q   + (size_t)(b * H_ + h) * L_ * E_;
    const float* kptr = k   + (size_t)(b * H_ + h) * E_ * L_;
    const float* vptr = v   + (size_t)(b * H_ + h) * L_ * E_;
    float*       optr = out + (size_t)(b * H_ + h) * L_ * E_;
    float*       wptr = wts + (size_t)(b * H_ + h) * (size_t)L_ * L_;
    const unsigned char* mptr = amask + (size_t)b * L_ * L_;

    // ---- Q tile as 16 A-fragments (16x4 f32 layout), pre-scaled by `scale`
    // (algebraically identical to scaling the scores; saves 8 v_mul/tile).
    // lanes 0-15 hold M=lane, lanes 16-31 hold M=lane-16;
    // VGPR0 = K = 4c+2*half, VGPR1 = K = 4c+1+2*half (contiguous pair).
    const int arow = qrow0 + n;
    v2f aq[16];
#pragma unroll
    for (int ch = 0; ch < 16; ++ch) {
        v2f t = *(const v2f*)(qptr + (size_t)arow * E_ + 4 * ch + 2 * half);
        t.x *= scale; t.y *= scale;
        aq[ch] = t;
    }

    // Finite mask sentinel: exp underflows to exactly 0, and no inf-inf NaNs
    // can appear in the online-softmax update.
    const float MASKVAL = -3.0e38f;

    // One 16x16 score tile (masked) in C/D layout: lane holds column n,
    // VGPR vv holds row vv + 8*half. Fully branchless masking (v_cndmask).
    auto score_tile = [&](int j16) -> v8f {
        v8f c = {0.f, 0.f, 0.f, 0.f, 0.f, 0.f, 0.f, 0.f};
#pragma unroll
        for (int ch = 0; ch < 16; ++ch) {
            const float* p = kptr + (size_t)(4 * ch + 2 * half) * L_ + j16 + n;
            v2f bk; bk.x = p[0]; bk.y = p[L_];   // B 4x16: VGPRr = rows r, r+2
            c = __builtin_amdgcn_wmma_f32_16x16x4_f32(
                    false, aq[ch], false, bk, (short)0, c, false, false);
        }
        const int col = j16 + n;
#pragma unroll
        for (int vv = 0; vv < 8; ++vv) {
            const int row = qrow0 + vv + 8 * half;
            const int mb  = (int)mptr[(size_t)row * L_ + col];
            const bool m  = ((mb != 0) | (csign * (col - row) > 0));
            c[vv] = m ? MASKVAL : c[vv];
        }
        return c;
    };

    // Tiles outside [jlo,jhi] are fully masked -> skipped (weights = 0).
    const int jlo = causal ? 0  : qb;
    const int jhi = causal ? qb : (NTJ - 1);

    // ---------------- pass 1: per-lane online max / sum ----------------
    float ml[8], sl[8];
#pragma unroll
    for (int vv = 0; vv < 8; ++vv) { ml[vv] = MASKVAL; sl[vv] = 0.f; }

    for (int j = jlo; j <= jhi; ++j) {
        v8f c = score_tile(j * 16);
#pragma unroll
        for (int vv = 0; vv < 8; ++vv) {
            float mn = fmaxf(ml[vv], c[vv]);
            sl[vv] = sl[vv] * __expf(ml[vv] - mn) + __expf(c[vv] - mn);
            ml[vv] = mn;
        }
    }

    // Fold the 16 lanes of each row (xor masks 1..8 stay within a half-wave).
    float M[8], invS[8];
#pragma unroll
    for (int vv = 0; vv < 8; ++vv) {
        float m = ml[vv];
#pragma unroll
        for (int off = 1; off < 16; off <<= 1)
            m = fmaxf(m, __shfl_xor(m, off, 32));
        float s = sl[vv] * __expf(ml[vv] - m);   // rescale lane-local sums
#pragma unroll
        for (int off = 1; off < 16; off <<= 1)
            s += __shfl_xor(s, off, 32);
        M[vv] = m;
        invS[vv] = 1.f / s;
    }

    // ---------------- pass 2: weights out + O = W @ V ----------------
    v8f o[4];
#pragma unroll
    for (int ec = 0; ec < 4; ++ec)
        o[ec] = (v8f){0.f, 0.f, 0.f, 0.f, 0.f, 0.f, 0.f, 0.f};

    float* wl = &wlds[wave][0];

    for (int j = jlo; j <= jhi; ++j) {
        const int j16 = j * 16;
        v8f c = score_tile(j16);

        float w8[8];
#pragma unroll
        for (int vv = 0; vv < 8; ++vv)
            w8[vv] = __expf(c[vv] - M[vv]) * invS[vv];

        // C-layout -> LDS (row-major tile). Same-wave LDS ops complete in
        // order (DScnt), so the store->load RAW needs no barrier.
#pragma unroll
        for (int vv = 0; vv < 8; ++vv)
            wl[(vv + 8 * half) * PITCH + n] = w8[vv];

        // Coalesced b128 weight stores: read tile rows back from LDS,
        // 2 x float4 per lane covers the 16x16 tile (dominant HBM stream).
#pragma unroll
        for (int t = 0; t < 2; ++t) {
            const int idx = lane + 32 * t;
            const int r   = idx >> 2;
            const int c4  = idx & 3;
            v4f wv = *(const v4f*)&wl[r * PITCH + c4 * 4];
            *(v4f*)&wptr[(size_t)(qrow0 + r) * L_ + j16 + c4 * 4] = wv;
        }

        // W tile as A-fragments (b64 reads, 8B aligned since PITCH%4==0).
        v2f aw[4];
#pragma unroll
        for (int kc = 0; kc < 4; ++kc)
            aw[kc] = *(const v2f*)&wl[n * PITCH + kc * 4 + 2 * half];

        // O(16x64) += W(16x16) * V(16x64): 4 e-chunks x 4 K-steps of WMMA.
#pragma unroll
        for (int ec = 0; ec < 4; ++ec) {
#pragma unroll
            for (int kc = 0; kc < 4; ++kc) {
                const float* p =
                    vptr + (size_t)(j16 + 4 * kc + 2 * half) * E_ + ec * 16 + n;
                v2f bv; bv.x = p[0]; bv.y = p[E_];
                o[ec] = __builtin_amdgcn_wmma_f32_16x16x4_f32(
                            false, aw[kc], false, bv, (short)0, o[ec],
                            false, false);
            }
        }
    }

    // Zero-fill the fully-masked weight tiles (softmax there is exactly 0).
    const int zlo = causal ? (qb + 1) : 0;
    const int zhi = causal ? (NTJ - 1) : (qb - 1);
    for (int j = zlo; j <= zhi; ++j) {
#pragma unroll
        for (int t = 0; t < 2; ++t) {          // 64 float4 per tile, 2/lane
            const int idx = lane + 32 * t;
            const int r   = idx >> 2;
            const int c4  = idx & 3;
            v4f z = {0.f, 0.f, 0.f, 0.f};
            *(v4f*)&wptr[(size_t)(qrow0 + r) * L_ + j * 16 + c4 * 4] = z;
        }
    }

    // Write O (already normalized since W was normalized before PV).
#pragma unroll
    for (int ec = 0; ec < 4; ++ec)
#pragma unroll
        for (int vv = 0; vv < 8; ++vv)
            optr[(size_t)(qrow0 + vv + 8 * half) * E_ + ec * 16 + n] = o[ec][vv];
}

extern "C" void kernel_launch(void* const* d_in, const int* in_sizes, int n_in,
                              void* d_out, int out_size, void* d_ws, size_t ws_size,
                              hipStream_t stream) {
    const float*         q  = (const float*)d_in[0];
    const float*         k  = (const float*)d_in[1];
    const float*         v  = (const float*)d_in[2];
    const unsigned char* am = (const unsigned char*)d_in[3];  // bool mask
    const float*         sc = (const float*)d_in[4];

    float* out = (float*)d_out;                          // (B,H,L,E)
    float* wts = out + (size_t)B_ * H_ * L_ * E_;        // (B,H,L,L)

    // 4096 wave-tasks (B*H*L/16), 4 waves (128 threads) per block.
    const int nblocks = (B_ * H_ * NTJ) / WAVES;         // 1024
    attn_dualmask_wmma<<<nblocks, 32 * WAVES, 0, stream>>>(q, k, v, am, sc,
                                                           out, wts);
}